// SEBlock_50156628082926
// MI455X (gfx1250) — compile-verified
//
#include <hip/hip_runtime.h>

// SE block: x[B=32,C=256,H=128,W=128] fp32
//   y = mean_{h,w} x                      [32,256]
//   z = relu(y @ w1^T)   w1:[16,256]      [32,16]
//   s = sigmoid(z @ w2^T) w2:[256,16]     [32,256]
//   out = x * s (broadcast)
//
// Memory-bound: 2 reads + 1 write of 537MB => ~1.61GB => ~69us @ 23.3TB/s.
// Excite stage uses V_WMMA_F32_16X16X4_F32 (fp32 matrix pipe, exact precision).
// Phase 3 walks planes in REVERSE so its first reads hit the ~192MB of x that
// phase 1 left in L2 (phase 1 streamed ascending; LRU keeps the tail).

#define Bdim 32
#define Cdim 256
#define HWdim (128 * 128)   // 16384 elements per (b,c) plane
#define BC    (Bdim * Cdim) // 8192 planes

typedef float v2f __attribute__((ext_vector_type(2)));
typedef float v4f __attribute__((ext_vector_type(4)));
typedef float v8f __attribute__((ext_vector_type(8)));

// ---------------------------------------------------------------------------
// Phase 1: squeeze. One block per (b,c) plane. 256 threads x 16 b128 loads
// = 64KB coalesced per block. Default (RT) temporal hint so the tail of x
// stays resident in the 192MB L2 for phase 3's reversed walk.
// ---------------------------------------------------------------------------
__global__ __launch_bounds__(256) void se_squeeze(const float* __restrict__ x,
                                                  float* __restrict__ y) {
    const int bc = blockIdx.x;
    const v4f* xp = reinterpret_cast<const v4f*>(x) + (size_t)bc * (HWdim / 4);

    float acc = 0.f;
#pragma unroll
    for (int i = 0; i < (HWdim / 4) / 256; ++i) {
        v4f v = xp[i * 256 + threadIdx.x];
        acc += v.x + v.y + v.z + v.w;
    }

    __shared__ float red[256];
    red[threadIdx.x] = acc;
    __syncthreads();
#pragma unroll
    for (int s = 128; s > 0; s >>= 1) {
        if (threadIdx.x < s) red[threadIdx.x] += red[threadIdx.x + s];
        __syncthreads();
    }
    if (threadIdx.x == 0) y[bc] = red[0] * (1.0f / HWdim);
}

// ---------------------------------------------------------------------------
// Phase 2: excite. Single 256-thread block (8 wave32).
// GEMM1: Z[32,16] = relu(Y[32,256] @ W1^T), K=256 -> 64 WMMA f32 16x16x4 per
//        M-tile. All 8 waves compute (wave w does M-tile w&1, redundantly) so
//        every wave issuing WMMA has EXEC == all ones (ISA 7.12 restriction).
// GEMM2: S[32,256] = sigmoid(Z @ W2^T), 2x16 output tiles, 4 tiles per wave,
//        K=16 -> 4 WMMA per tile.
// Layouts per CDNA5 ISA 7.12.2:
//   A (16x4 f32, 2 VGPR): lanes 0-15 -> M=lane, K={0,1}; lanes 16-31 -> K={2,3}
//   B (4x16 f32, 2 VGPR): mirrored (lanes 0-15 N=lane K={0,1}; 16-31 K={2,3})
//   C/D (16x16 f32, 8 VGPR): VGPR v holds M=v (lanes 0-15) / M=v+8 (16-31)
// ---------------------------------------------------------------------------
__global__ __launch_bounds__(256) void se_excite(const float* __restrict__ y,   // [32,256]
                                                 const float* __restrict__ w1,  // [16,256]
                                                 const float* __restrict__ w2,  // [256,16]
                                                 float* __restrict__ s) {       // [32,256]
    __shared__ float zsh[32][17];  // padded to dodge bank conflicts

    const int lane  = threadIdx.x & 31;
    const int wid   = threadIdx.x >> 5;    // 0..7
    const int n16   = lane & 15;           // N (or M for A loads) index
    const int khalf = (lane >> 4) * 2;     // 0 for lanes 0-15, 2 for 16-31
    const int mhi   = (lane >> 4) * 8;     // C/D row offset for hi half-wave

    // ---- GEMM1: Z = relu(Y @ W1^T) ----
    {
        const int mt = wid & 1;  // M-tile, computed redundantly by 4 waves each
        v8f acc = {};
        for (int kb = 0; kb < 256 / 4; ++kb) {
            const int k0 = kb * 4 + khalf;
            v2f a, b;
            a.x = y[(mt * 16 + n16) * 256 + k0];
            a.y = y[(mt * 16 + n16) * 256 + k0 + 1];
            b.x = w1[n16 * 256 + k0];       // B[k][n] = w1[n][k]
            b.y = w1[n16 * 256 + k0 + 1];
            acc = __builtin_amdgcn_wmma_f32_16x16x4_f32(
                false, a, false, b, (short)0, acc, false, false);
        }
#pragma unroll
        for (int v = 0; v < 8; ++v) {
            float z = acc[v];
            zsh[mt * 16 + mhi + v][n16] = z > 0.f ? z : 0.f;  // redundant same-value writes
        }
    }
    __syncthreads();

    // ---- GEMM2: S = sigmoid(Z @ W2^T) ----
#pragma unroll
    for (int t = 0; t < 4; ++t) {
        const int tile = wid * 4 + t;  // 0..31
        const int mt = tile >> 4;      // 0..1
        const int nt = tile & 15;      // 0..15
        v8f acc = {};
#pragma unroll
        for (int kb = 0; kb < 4; ++kb) {
            const int k0 = kb * 4 + khalf;
            v2f a, b;
            a.x = zsh[mt * 16 + n16][k0];
            a.y = zsh[mt * 16 + n16][k0 + 1];
            b.x = w2[(nt * 16 + n16) * 16 + k0];   // B[k][n] = w2[nt*16+n][k]
            b.y = w2[(nt * 16 + n16) * 16 + k0 + 1];
            acc = __builtin_amdgcn_wmma_f32_16x16x4_f32(
                false, a, false, b, (short)0, acc, false, false);
        }
#pragma unroll
        for (int v = 0; v < 8; ++v) {
            float d = acc[v];
            s[(mt * 16 + mhi + v) * 256 + nt * 16 + n16] = 1.0f / (1.0f + __expf(-d));
        }
    }
}

// ---------------------------------------------------------------------------
// Phase 3: scale. One block per (b,c) plane, walked in REVERSE plane order so
// the first blocks hit the x-tail that phase 1 left in L2. s[bc] is block-
// uniform (scalar load). NT hints: hits still hit; misses don't pollute.
// ---------------------------------------------------------------------------
__global__ __launch_bounds__(256) void se_scale(const float* __restrict__ x,
                                                const float* __restrict__ s,
                                                float* __restrict__ out) {
    const int bc = (BC - 1) - blockIdx.x;   // reverse plane order for L2 reuse
    const float sv = s[bc];
    const v4f* xp = reinterpret_cast<const v4f*>(x) + (size_t)bc * (HWdim / 4);
    v4f* op = reinterpret_cast<v4f*>(out) + (size_t)bc * (HWdim / 4);

#pragma unroll
    for (int i = 0; i < (HWdim / 4) / 256; ++i) {
        const int idx = i * 256 + threadIdx.x;
        v4f v = __builtin_nontemporal_load(&xp[idx]);
        v4f r;
        r.x = v.x * sv; r.y = v.y * sv; r.z = v.z * sv; r.w = v.w * sv;
        __builtin_nontemporal_store(r, &op[idx]);
    }
}

// ---------------------------------------------------------------------------
extern "C" void kernel_launch(void* const* d_in, const int* in_sizes, int n_in,
                              void* d_out, int out_size, void* d_ws, size_t ws_size,
                              hipStream_t stream) {
    const float* x  = (const float*)d_in[0];  // [32,256,128,128]
    const float* w1 = (const float*)d_in[1];  // [16,256]
    const float* w2 = (const float*)d_in[2];  // [256,16]
    float* out = (float*)d_out;

    float* y = (float*)d_ws;     // [32,256] pooled means
    float* s = y + BC;           // [32,256] sigmoid scales

    se_squeeze<<<BC, 256, 0, stream>>>(x, y);
    se_excite<<<1, 256, 0, stream>>>(y, w1, w2, s);
    se_scale<<<BC, 256, 0, stream>>>(x, s, out);
}